// OffsetHead_69200513073480
// MI455X (gfx1250) — compile-verified
//
#include <hip/hip_runtime.h>
#include <math.h>

typedef float v2f __attribute__((ext_vector_type(2)));
typedef float v8f __attribute__((ext_vector_type(8)));
typedef unsigned long long u64;

#define CIN    64
#define DDESC  16
#define SHIFT_ 4096
#define MBASE  16384ULL
#define HBITS  21
#define HSIZE  (1u << HBITS)         // 2,097,152 slots (~0.48 load factor)
#define HMASK  (HSIZE - 1u)
#define EMPTY_KEY 0xFFFFFFFFFFFFFFFFULL
#define SCAN_CHUNK 2048              // slots per block in compaction
#define SCAN_NBLK  (HSIZE / SCAN_CHUNK)  // 1024

__device__ __forceinline__ unsigned hash_u64(u64 key) {
  key *= 0x9E3779B97F4A7C15ULL;
  return (unsigned)(key >> 43) & HMASK;
}

// round(sign(F) * expm1(|F|)) with round-half-to-even, matching jnp.round
__device__ __forceinline__ int off_round(float f) {
  return (int)rintf(copysignf(expm1f(fabsf(f)), f));
}

// ---------------------------------------------------------------------------
// 0) Workspace init (re-run every call: harness does not re-poison buffers)
// ---------------------------------------------------------------------------
__global__ __launch_bounds__(256) void init_ws(
    float* __restrict__ descsum, u64* __restrict__ coordsum,
    int* __restrict__ counts, u64* __restrict__ hashk, int* __restrict__ hashv,
    long long n)
{
  long long i = (long long)blockIdx.x * blockDim.x + threadIdx.x;
  long long total = 16LL * n;                 // dominant range (descsum)
  if (i >= total) return;
  descsum[i] = 0.0f;
  if (i < 4LL * n) coordsum[i] = 0ULL;
  if (i < n)       counts[i] = 0;
  if (i < (long long)HSIZE) { hashk[i] = EMPTY_KEY; hashv[i] = -1; }
}

// ---------------------------------------------------------------------------
// 1) Dual-head GEMM via fp32 WMMA. One wave32 per 16-point tile.
//    A: 16x4 f32 per step (lane = M, half-wave picks K pair, ISA 7.12.2)
//    B: W_dec zero-padded to 16 cols; W_desc exactly 16 cols.
//    K=64 -> 16 accumulation steps x 2 heads = 32 v_wmma per tile.
// ---------------------------------------------------------------------------
__global__ __launch_bounds__(256) void gemm_heads_wmma(
    const float* __restrict__ feats, const unsigned char* __restrict__ mask,
    const float* __restrict__ Wdec, const float* __restrict__ bdec,
    const float* __restrict__ Wdesc, const float* __restrict__ bdesc,
    float* __restrict__ out_off, float* __restrict__ out_desc, int n)
{
  const int lane  = threadIdx.x & 31;
  const int wv    = (blockIdx.x << 3) | (threadIdx.x >> 5);  // 8 waves / block
  const int half  = lane >> 4;        // K-pair selector for A/B
  const int nm    = lane & 15;        // M for A, N for B/C/D
  const int tile0 = wv << 4;
  if (tile0 >= n) return;             // wave-uniform exit (EXEC stays full)

  const int arow = tile0 + nm;
  const int arc  = arow < n ? arow : n - 1;
  const float mval = (arow < n && mask[arc] != 0) ? 1.0f : 0.0f;
  const float* fr = feats + (long long)arc * CIN;

  v8f accO = {0.f,0.f,0.f,0.f,0.f,0.f,0.f,0.f};
  v8f accD = {0.f,0.f,0.f,0.f,0.f,0.f,0.f,0.f};

  const float m3 = (nm < 3) ? 1.0f : 0.0f;   // zero-pad W_dec columns 3..15
  const int   n3 = (nm < 3) ? nm : 0;

  #pragma unroll
  for (int s = 0; s < 16; ++s) {
    const int k = (s << 2) + (half << 1);
    v2f a;  a.x  = fr[k] * mval;              a.y  = fr[k + 1] * mval;
    v2f bo; bo.x = Wdec[k*3 + n3] * m3;       bo.y = Wdec[(k+1)*3 + n3] * m3;
    v2f bd; bd.x = Wdesc[k*DDESC + nm];       bd.y = Wdesc[(k+1)*DDESC + nm];
    accO = __builtin_amdgcn_wmma_f32_16x16x4_f32(false, a, false, bo, (short)0, accO, false, false);
    accD = __builtin_amdgcn_wmma_f32_16x16x4_f32(false, a, false, bd, (short)0, accD, false, false);
  }

  const float bias_o = (nm < 3) ? bdec[n3] : 0.0f;
  const float bias_d = bdesc[nm];
  #pragma unroll
  for (int r = 0; r < 8; ++r) {
    const int row = tile0 + r + (half << 3);   // C/D layout: VGPR r -> M=r(+8)
    if (row < n) {
      if (nm < 3) out_off[(long long)row * 3 + nm] = accO[r] + bias_o;
      out_desc[(long long)row * DDESC + nm] = accD[r] + bias_d;  // raw, staged
    }
  }
}

// ---------------------------------------------------------------------------
// 2) Build packed voxel keys and insert into open-addressing hash table.
// ---------------------------------------------------------------------------
__global__ __launch_bounds__(256) void build_keys_insert(
    const float* __restrict__ out_off, const int* __restrict__ coords,
    u64* __restrict__ keys, u64* __restrict__ hashk, int n)
{
  const int i = blockIdx.x * blockDim.x + threadIdx.x;
  if (i >= n) return;
  const float* o = out_off + (long long)i * 3;
  const u64 c0 = (u64)(long long)(coords[i*4 + 0] + SHIFT_);
  const u64 c1 = (u64)(long long)(coords[i*4 + 1] + off_round(o[0]) + SHIFT_);
  const u64 c2 = (u64)(long long)(coords[i*4 + 2] + off_round(o[1]) + SHIFT_);
  const u64 c3 = (u64)(long long)(coords[i*4 + 3] + off_round(o[2]) + SHIFT_);
  const u64 key = ((c0 * MBASE + c1) * MBASE + c2) * MBASE + c3;
  keys[i] = key;
  unsigned h = hash_u64(key);
  for (;;) {
    u64 prev = atomicCAS(&hashk[h], EMPTY_KEY, key);
    if (prev == EMPTY_KEY || prev == key) break;
    h = (h + 1) & HMASK;
  }
}

// ---------------------------------------------------------------------------
// 3) Deterministic voxel-id compaction over hash slots (count / scan / assign)
// ---------------------------------------------------------------------------
__global__ __launch_bounds__(256) void count_occ(
    const u64* __restrict__ hashk, int* __restrict__ blocksums)
{
  __shared__ int sdata[256];
  const int t = threadIdx.x;
  const int base = blockIdx.x * SCAN_CHUNK + t * 8;
  int cnt = 0;
  #pragma unroll
  for (int j = 0; j < 8; ++j) cnt += (hashk[base + j] != EMPTY_KEY) ? 1 : 0;
  sdata[t] = cnt; __syncthreads();
  for (int off = 128; off > 0; off >>= 1) {
    if (t < off) sdata[t] += sdata[t + off];
    __syncthreads();
  }
  if (t == 0) blocksums[blockIdx.x] = sdata[0];
}

__global__ void scan_blocks(int* __restrict__ blocksums, int nblk)
{
  if (blockIdx.x == 0 && threadIdx.x == 0) {
    int run = 0;
    for (int i = 0; i < nblk; ++i) { int v = blocksums[i]; blocksums[i] = run; run += v; }
  }
}

__global__ __launch_bounds__(256) void assign_ids(
    const u64* __restrict__ hashk, int* __restrict__ hashv,
    const int* __restrict__ blockoffs)
{
  __shared__ int sdata[256];
  const int t = threadIdx.x;
  const int base = blockIdx.x * SCAN_CHUNK + t * 8;
  int cnt = 0;
  #pragma unroll
  for (int j = 0; j < 8; ++j) cnt += (hashk[base + j] != EMPTY_KEY) ? 1 : 0;
  sdata[t] = cnt; __syncthreads();
  for (int off = 1; off < 256; off <<= 1) {        // inclusive Hillis-Steele
    int v = (t >= off) ? sdata[t - off] : 0;
    __syncthreads();
    sdata[t] += v;
    __syncthreads();
  }
  int id = blockoffs[blockIdx.x] + (sdata[t] - cnt);
  for (int j = 0; j < 8; ++j)
    if (hashk[base + j] != EMPTY_KEY) hashv[base + j] = id++;
}

// ---------------------------------------------------------------------------
// 4) Per-point lookup: inv map, counts (int), coord sums (int64 atomics)
// ---------------------------------------------------------------------------
__global__ __launch_bounds__(256) void lookup_accum(
    const u64* __restrict__ keys, const u64* __restrict__ hashk,
    const int* __restrict__ hashv, int* __restrict__ counts,
    u64* __restrict__ coordsum, int* __restrict__ invw,
    float* __restrict__ out_inv, int n)
{
  const int i = blockIdx.x * blockDim.x + threadIdx.x;
  if (i >= n) return;
  const u64 key = keys[i];
  unsigned h = hash_u64(key);
  while (hashk[h] != key) h = (h + 1) & HMASK;
  const int v = hashv[h];
  invw[i] = v;
  out_inv[i] = (float)v;
  atomicAdd(&counts[v], 1);
  const long long c3 = (long long)( key        & (MBASE - 1)) - SHIFT_;
  const long long c2 = (long long)((key >> 14) & (MBASE - 1)) - SHIFT_;
  const long long c1 = (long long)((key >> 28) & (MBASE - 1)) - SHIFT_;
  const long long c0 = (long long)((key >> 42) & (MBASE - 1)) - SHIFT_;
  atomicAdd(&coordsum[(long long)v*4 + 0], (u64)c0);
  atomicAdd(&coordsum[(long long)v*4 + 1], (u64)c1);
  atomicAdd(&coordsum[(long long)v*4 + 2], (u64)c2);
  atomicAdd(&coordsum[(long long)v*4 + 3], (u64)c3);
}

// 5) Descriptor segment sums: one thread per (point, channel)
__global__ __launch_bounds__(256) void desc_accum(
    const float* __restrict__ desc_raw, const int* __restrict__ invw,
    float* __restrict__ descsum, long long total)
{
  const long long idx = (long long)blockIdx.x * blockDim.x + threadIdx.x;
  if (idx >= total) return;
  const int i = (int)(idx >> 4), c = (int)(idx & 15);
  atomicAdd(&descsum[(long long)invw[i] * DDESC + c], desc_raw[idx]);
}

// ---------------------------------------------------------------------------
// 6) Per-voxel finalize: mean coords (truncate to int), L2-normalized desc
// ---------------------------------------------------------------------------
__global__ __launch_bounds__(256) void finalize_voxels(
    const int* __restrict__ counts, const u64* __restrict__ coordsum,
    const float* __restrict__ descsum, float* __restrict__ out_coords,
    float* __restrict__ out_desc, int n)
{
  const int v = blockIdx.x * blockDim.x + threadIdx.x;
  if (v >= n) return;
  const int c = counts[v];
  const float denom = (float)(c > 1 ? c : 1);
  #pragma unroll
  for (int j = 0; j < 4; ++j) {
    const long long s = (long long)coordsum[(long long)v*4 + j];
    out_coords[(long long)v*4 + j] = (float)(int)((float)s / denom);
  }
  float d[DDESC]; float ss = 0.0f;
  #pragma unroll
  for (int k = 0; k < DDESC; ++k) {
    d[k] = descsum[(long long)v*DDESC + k] / denom;
    ss += d[k] * d[k];
  }
  const float scale = 1.0f / fmaxf(sqrtf(ss), 1e-12f);
  #pragma unroll
  for (int k = 0; k < DDESC; ++k)
    out_desc[(long long)v*DDESC + k] = d[k] * scale;
}

// 7) Per-point scores: log1p(count[inv])
__global__ __launch_bounds__(256) void scores_kernel(
    const int* __restrict__ counts, const int* __restrict__ invw,
    float* __restrict__ out_scores, int n)
{
  const int i = blockIdx.x * blockDim.x + threadIdx.x;
  if (i >= n) return;
  out_scores[i] = log1pf((float)counts[invw[i]]);
}

// ---------------------------------------------------------------------------
extern "C" void kernel_launch(void* const* d_in, const int* in_sizes, int n_in,
                              void* d_out, int out_size, void* d_ws, size_t ws_size,
                              hipStream_t stream)
{
  const float*         feats  = (const float*)d_in[0];
  const int*           coords = (const int*)d_in[1];
  const unsigned char* mask   = (const unsigned char*)d_in[2];
  const float*         Wdec   = (const float*)d_in[3];
  const float*         bdec   = (const float*)d_in[4];
  const float*         Wdesc  = (const float*)d_in[5];
  const float*         bdesc  = (const float*)d_in[6];

  const long long n = in_sizes[2];          // mask has N elements

  // d_out: [offsets 3N | desc_norm 16N | out_coords 4N | out_scores N | inv N]
  float* out        = (float*)d_out;
  float* out_off    = out;
  float* out_desc   = out + 3  * n;         // staged raw desc, finalized later
  float* out_coords = out + 19 * n;
  float* out_scores = out + 23 * n;
  float* out_inv    = out + 24 * n;

  // workspace carve-up (~137 MB)
  char* w = (char*)d_ws;
  float* descsum  = (float*)w;  w += 16 * n * sizeof(float);
  u64*   coordsum = (u64*)w;    w += 4  * n * sizeof(u64);
  int*   counts   = (int*)w;    w += n * sizeof(int);
  int*   invw     = (int*)w;    w += n * sizeof(int);
  u64*   keys     = (u64*)w;    w += n * sizeof(u64);
  u64*   hashk    = (u64*)w;    w += (size_t)HSIZE * sizeof(u64);
  int*   hashv    = (int*)w;    w += (size_t)HSIZE * sizeof(int);
  int*   blocksums= (int*)w;

  const int T = 256;
  const unsigned gInit  = (unsigned)((16 * n + T - 1) / T);
  const unsigned gPts   = (unsigned)((n + T - 1) / T);
  const long long waves = (n + 15) / 16;
  const unsigned gGemm  = (unsigned)((waves + 7) / 8);     // 8 waves / block
  const long long dtot  = 16 * n;
  const unsigned gDesc  = (unsigned)((dtot + T - 1) / T);

  init_ws<<<gInit, T, 0, stream>>>(descsum, coordsum, counts, hashk, hashv, n);
  gemm_heads_wmma<<<gGemm, T, 0, stream>>>(feats, mask, Wdec, bdec, Wdesc, bdesc,
                                           out_off, out_desc, (int)n);
  build_keys_insert<<<gPts, T, 0, stream>>>(out_off, coords, keys, hashk, (int)n);
  count_occ<<<SCAN_NBLK, T, 0, stream>>>(hashk, blocksums);
  scan_blocks<<<1, 32, 0, stream>>>(blocksums, SCAN_NBLK);
  assign_ids<<<SCAN_NBLK, T, 0, stream>>>(hashk, hashv, blocksums);
  lookup_accum<<<gPts, T, 0, stream>>>(keys, hashk, hashv, counts, coordsum,
                                       invw, out_inv, (int)n);
  desc_accum<<<gDesc, T, 0, stream>>>(out_desc, invw, descsum, dtot);
  finalize_voxels<<<gPts, T, 0, stream>>>(counts, coordsum, descsum,
                                          out_coords, out_desc, (int)n);
  scores_kernel<<<gPts, T, 0, stream>>>(counts, invw, out_scores, (int)n);
}